// RegionProposalNetwork_64312840290699
// MI455X (gfx1250) — compile-verified
//
#include <hip/hip_runtime.h>
#include <hip/hip_bf16.h>

typedef __attribute__((ext_vector_type(2))) float v2f;
typedef __attribute__((ext_vector_type(8))) float v8f;

#define CIN   512
#define HF    50
#define WF    76
#define MTOT  3800      // 50*76
#define ATOT  34200     // MTOT*9
#define NSORT 65536
#define NPRE  6000
#define NPOST 300

// output layout (floats): rpn_locs[136800] | rpn_scores[68400] | rois[1200] | anchor[136800]
#define OFF_LOCS   0
#define OFF_SCORES 136800
#define OFF_ROIS   205200
#define OFF_ANCHOR 206400

// workspace layout (bytes)
#define WT_OFF   0u                    // 9*512*512 f32 = 9437184 B
#define X_OFF    9437184u              // 3800*512 f32  = 7782400 B
#define BOX_OFF  17219584u             // 34200 float4  =  547200 B
#define KEY_OFF  17766784u             // 65536 u64     =  524288 B
#define BK_OFF   18291072u             // 6000 float4   =   96000 B
#define TS_OFF   18387072u             // 6000 f32      =   24000 B
#define KEEP_OFF 18411072u             // 6000 u32      =   24000 B

__device__ __forceinline__ unsigned f2ord(float f) {
    unsigned u = __float_as_uint(f);
    return (u & 0x80000000u) ? ~u : (u | 0x80000000u);
}
__device__ __forceinline__ float ord2f(unsigned u) {
    return (u & 0x80000000u) ? __uint_as_float(u ^ 0x80000000u) : __uint_as_float(~u);
}

// ---------------- kernel 0: weight transpose OIHW -> [tap][ci][co] ----------
__global__ void k_transpose_w(const float* __restrict__ w, float* __restrict__ wt) {
    int i = blockIdx.x * 256 + threadIdx.x;          // over output index
    if (i >= 9 * CIN * CIN) return;
    int tap = i / (CIN * CIN);
    int rem = i - tap * (CIN * CIN);
    int ci  = rem >> 9;
    int co  = rem & 511;
    wt[i] = w[co * (CIN * 9) + ci * 9 + tap];
}

// ---------------- kernel 1: conv3x3 + ReLU via f32 WMMA implicit GEMM -------
// Block tile 128(M) x 64(N); 8 waves in 4(M) x 2(N); each wave 32x32 via 4 accums.
// K = 4608 staged through double-buffered LDS in 16-wide chunks (288 stages).
// B tiles stored pair-major ([kp][col*2+slot], rows padded to 160 floats) so a
// fragment is one aligned, bank-conflict-free ds_load_b64.
__global__ __launch_bounds__(256)
void k_conv3x3_wmma(const float* __restrict__ feat, const float* __restrict__ wt,
                    const float* __restrict__ bias, float* __restrict__ xbuf) {
    __shared__ float As[2][128][20];   // pad 20: conflict-free b64 frag loads
    __shared__ float Bs[2][8][160];    // pair-major; stride 160 => 32-bank rotation/row

    const int tid  = threadIdx.x;
    const int lane = tid & 31;
    const int wave = tid >> 5;
    const int wm   = wave & 3;         // 4 waves along M (32 rows each)
    const int wn   = wave >> 2;        // 2 waves along N (32 cols each)
    const int half = lane >> 4;
    const int l15  = lane & 15;
    const int mtile = blockIdx.x * 128;
    const int ntile = blockIdx.y * 64;

    // loop-invariant per-thread staging coordinates (hoists the /76, %76 chains)
    int aY[8], aX[8];
    bool aValid[8];
    #pragma unroll
    for (int it = 0; it < 8; ++it) {
        int t  = tid + it * 256;
        int mi = t & 127;
        int m  = mtile + mi;
        aValid[it] = (m < MTOT);
        int mm = aValid[it] ? m : 0;
        aY[it] = mm / WF;
        aX[it] = mm % WF;
    }

    v8f acc00 = {0.f,0.f,0.f,0.f,0.f,0.f,0.f,0.f};
    v8f acc01 = {0.f,0.f,0.f,0.f,0.f,0.f,0.f,0.f};
    v8f acc10 = {0.f,0.f,0.f,0.f,0.f,0.f,0.f,0.f};
    v8f acc11 = {0.f,0.f,0.f,0.f,0.f,0.f,0.f,0.f};

    // ---- prologue: stage 0 into buffer 0 ----
    {
        const int cblock = 0;
        const int dy = -1, dx = -1;
        #pragma unroll
        for (int it = 0; it < 8; ++it) {
            int t = tid + it * 256;
            int kj = t >> 7;
            int yy = aY[it] + dy, xx = aX[it] + dx;
            float v = 0.f;
            if (aValid[it] && yy >= 0 && yy < HF && xx >= 0 && xx < WF)
                v = feat[(cblock + kj) * MTOT + yy * WF + xx];
            As[0][t & 127][kj] = v;
        }
        const float* bsrc = wt + ntile;   // tap 0, cblock 0
        #pragma unroll
        for (int it = 0; it < 4; ++it) {
            int t  = tid + it * 256;
            int ni = t & 63, kj = t >> 6;
            Bs[0][kj >> 1][ni * 2 + (kj & 1)] = bsrc[kj * CIN + ni];
        }
    }
    __syncthreads();

    const int colA = (wn * 32 + l15) * 2;       // pair-major column offset, subtile 0
    const int colB = colA + 32;                 // subtile 1 (col + 16)

    for (int s = 0; s < 288; ++s) {
        const int cur = s & 1, nxt = cur ^ 1;
        const bool havenext = (s + 1 < 288);

        // ---- issue global loads for stage s+1 into registers (overlaps WMMA) ----
        float aReg[8], bReg[4];
        if (havenext) {
            const int s1 = s + 1;
            const int tap = s1 >> 5;
            const int cblock = (s1 & 31) << 4;
            const int dy = tap / 3 - 1;
            const int dx = tap % 3 - 1;
            #pragma unroll
            for (int it = 0; it < 8; ++it) {
                int t = tid + it * 256;
                int kj = t >> 7;
                int yy = aY[it] + dy, xx = aX[it] + dx;
                float v = 0.f;
                if (aValid[it] && yy >= 0 && yy < HF && xx >= 0 && xx < WF)
                    v = feat[(cblock + kj) * MTOT + yy * WF + xx];
                aReg[it] = v;
            }
            const float* bsrc = wt + tap * (CIN * CIN) + cblock * CIN + ntile;
            #pragma unroll
            for (int it = 0; it < 4; ++it) {
                int t = tid + it * 256;
                bReg[it] = bsrc[(t >> 6) * CIN + (t & 63)];
            }
        }

        // ---- preload all 16 fragments, then 16 WMMAs ----
        v2f a0[4], a1[4], b0[4], b1[4];
        #pragma unroll
        for (int kx = 0; kx < 4; ++kx) {
            const int krow = kx * 4 + 2 * half;
            const int kp   = kx * 2 + half;
            a0[kx] = *(const v2f*)&As[cur][wm * 32 + l15][krow];
            a1[kx] = *(const v2f*)&As[cur][wm * 32 + 16 + l15][krow];
            b0[kx] = *(const v2f*)&Bs[cur][kp][colA];
            b1[kx] = *(const v2f*)&Bs[cur][kp][colB];
        }
        #pragma unroll
        for (int kx = 0; kx < 4; ++kx) {
            acc00 = __builtin_amdgcn_wmma_f32_16x16x4_f32(false, a0[kx], false, b0[kx],
                                                          (short)0, acc00, false, false);
            acc01 = __builtin_amdgcn_wmma_f32_16x16x4_f32(false, a0[kx], false, b1[kx],
                                                          (short)0, acc01, false, false);
            acc10 = __builtin_amdgcn_wmma_f32_16x16x4_f32(false, a1[kx], false, b0[kx],
                                                          (short)0, acc10, false, false);
            acc11 = __builtin_amdgcn_wmma_f32_16x16x4_f32(false, a1[kx], false, b1[kx],
                                                          (short)0, acc11, false, false);
        }

        // ---- commit staged registers to buffer `nxt` ----
        if (havenext) {
            #pragma unroll
            for (int it = 0; it < 8; ++it) {
                int t = tid + it * 256;
                As[nxt][t & 127][t >> 7] = aReg[it];
            }
            #pragma unroll
            for (int it = 0; it < 4; ++it) {
                int t  = tid + it * 256;
                int ni = t & 63, kj = t >> 6;
                Bs[nxt][kj >> 1][ni * 2 + (kj & 1)] = bReg[it];
            }
        }
        __syncthreads();
    }

    // ---- epilogue: bias + ReLU, store x as [m][co] (coalesced across lanes) ----
    const int co0 = ntile + wn * 32 + l15;
    const float bias0 = bias[co0];
    const float bias1 = bias[co0 + 16];
    #pragma unroll
    for (int i = 0; i < 8; ++i) {
        int m0 = mtile + wm * 32 + i + 8 * half;
        int m1 = m0 + 16;
        if (m0 < MTOT) {
            float v0 = acc00[i] + bias0; v0 = v0 > 0.f ? v0 : 0.f;
            float v1 = acc01[i] + bias1; v1 = v1 > 0.f ? v1 : 0.f;
            xbuf[m0 * CIN + co0]      = v0;
            xbuf[m0 * CIN + co0 + 16] = v1;
        }
        if (m1 < MTOT) {
            float v0 = acc10[i] + bias0; v0 = v0 > 0.f ? v0 : 0.f;
            float v1 = acc11[i] + bias1; v1 = v1 > 0.f ? v1 : 0.f;
            xbuf[m1 * CIN + co0]      = v0;
            xbuf[m1 * CIN + co0 + 16] = v1;
        }
    }
}

// ---------------- kernel 2: 1x1 convs (loc 36ch + score 18ch) ---------------
__global__ void k_conv1x1(const float* __restrict__ xbuf,
                          const float* __restrict__ loc_w, const float* __restrict__ loc_b,
                          const float* __restrict__ score_w, const float* __restrict__ score_b,
                          float* __restrict__ out) {
    const int m = blockIdx.x;
    const int n = threadIdx.x;
    if (n >= 54) return;
    const float* wv; float bias;
    if (n < 36) { wv = loc_w + n * CIN;          bias = loc_b[n]; }
    else        { wv = score_w + (n - 36) * CIN; bias = score_b[n - 36]; }
    const float4* x4 = (const float4*)(xbuf + m * CIN);
    const float4* w4 = (const float4*)wv;
    float acc = 0.f;
    #pragma unroll 4
    for (int c = 0; c < CIN / 4; ++c) {
        float4 a = x4[c], b = w4[c];
        acc += a.x * b.x + a.y * b.y + a.z * b.z + a.w * b.w;
    }
    acc += bias;
    if (n < 36) out[OFF_LOCS + m * 36 + n] = acc;           // (H,W,36) flat
    else        out[OFF_SCORES + m * 18 + (n - 36)] = acc;  // (H,W,18) flat
}

// -------- kernel 3: anchors, softmax-fg, loc2bbox, clip, filter, keys -------
__global__ void k_prep(const float* __restrict__ locs, const float* __restrict__ scores,
                       const int* __restrict__ imh, const int* __restrict__ imw,
                       float* __restrict__ anchorOut, float4* __restrict__ boxes,
                       unsigned long long* __restrict__ keys) {
    int i = blockIdx.x * 256 + threadIdx.x;
    if (i >= NSORT) return;
    if (i >= ATOT) { keys[i] = 0ull; return; }   // padding sorts to bottom

    int m = i / 9, a = i - m * 9;
    int y = m / WF, x = m - y * WF;
    const float ratios[3] = {0.5f, 1.f, 2.f};
    const float scales[3] = {8.f, 16.f, 32.f};
    float r = ratios[a / 3], s = scales[a % 3];
    float hh = 16.f * s * sqrtf(r);
    float ww = 16.f * s * sqrtf(1.f / r);
    float cy = y * 16.f + 8.f, cx = x * 16.f + 8.f;
    float ay1 = cy - 0.5f * hh, ax1 = cx - 0.5f * ww;
    float ay2 = cy + 0.5f * hh, ax2 = cx + 0.5f * ww;
    anchorOut[i * 4 + 0] = ay1; anchorOut[i * 4 + 1] = ax1;
    anchorOut[i * 4 + 2] = ay2; anchorOut[i * 4 + 3] = ax2;

    float dy = locs[m * 36 + a * 4 + 0];
    float dx = locs[m * 36 + a * 4 + 1];
    float dh = locs[m * 36 + a * 4 + 2];
    float dw = locs[m * 36 + a * 4 + 3];
    float ha = ay2 - ay1, wa = ax2 - ax1;
    float ncy = dy * ha + (ay1 + 0.5f * ha);
    float ncx = dx * wa + (ax1 + 0.5f * wa);
    float nh = expf(dh) * ha, nw = expf(dw) * wa;
    float Hc = (float)imh[0], Wc = (float)imw[0];
    float b0 = fminf(fmaxf(ncy - 0.5f * nh, 0.f), Hc);
    float b1 = fminf(fmaxf(ncx - 0.5f * nw, 0.f), Wc);
    float b2 = fminf(fmaxf(ncy + 0.5f * nh, 0.f), Hc);
    float b3 = fminf(fmaxf(ncx + 0.5f * nw, 0.f), Wc);
    boxes[i] = make_float4(b0, b1, b2, b3);

    float s0 = scores[m * 18 + a * 2], s1 = scores[m * 18 + a * 2 + 1];
    float fg = 1.f / (1.f + expf(s0 - s1));        // softmax[...,1]
    float hs = b2 - b0, ws2 = b3 - b1;
    float sc = (hs >= 16.f && ws2 >= 16.f) ? fg : -INFINITY;
    // key: (ordered score, ~index) -> ascending sort; ties resolve to lower index first
    keys[i] = ((unsigned long long)f2ord(sc) << 32) | (unsigned)(~(unsigned)i);
}

// ---------------- kernel 4: one bitonic compare/exchange pass ---------------
__global__ void k_bitonic(unsigned long long* __restrict__ keys, int j, int k) {
    int i = blockIdx.x * 256 + threadIdx.x;
    int ixj = i ^ j;
    if (ixj > i) {
        unsigned long long a = keys[i], b = keys[ixj];
        bool asc = (i & k) == 0;
        if (asc ? (a > b) : (a < b)) { keys[i] = b; keys[ixj] = a; }
    }
}

// ---------------- kernel 5: gather top-6000 (rank order), init keep ---------
__global__ void k_gather_top(const unsigned long long* __restrict__ keys,
                             const float4* __restrict__ boxes,
                             float4* __restrict__ bk, float* __restrict__ topscore,
                             unsigned* __restrict__ keepArr) {
    int r = blockIdx.x * 256 + threadIdx.x;
    if (r >= NPRE) return;
    unsigned long long kkey = keys[NSORT - 1 - r];      // ascending sort -> read from top
    unsigned idx = ~(unsigned)(kkey & 0xFFFFFFFFull);
    bk[r] = boxes[idx];
    topscore[r] = ord2f((unsigned)(kkey >> 32));
    keepArr[r] = 1u;
}

// ---------------- kernel 6: greedy NMS (serial-i, parallel-j) ---------------
__global__ void k_nms(const float4* __restrict__ bk, unsigned* __restrict__ keepArr) {
    extern __shared__ float4 sbox[];                  // 6000 * 16B = 96 KB (WGP LDS 320 KB)
    __shared__ unsigned keepBits[(NPRE + 31) / 32];
    __shared__ int curKeep;
    const int tid = threadIdx.x;
    for (int j = tid; j < NPRE; j += blockDim.x) sbox[j] = bk[j];
    for (int w = tid; w < (NPRE + 31) / 32; w += blockDim.x) keepBits[w] = 0xFFFFFFFFu;
    __syncthreads();
    for (int i = 0; i < NPRE; ++i) {
        if (tid == 0) curKeep = (keepBits[i >> 5] >> (i & 31)) & 1;
        __syncthreads();
        if (curKeep) {
            float4 bi = sbox[i];
            float areaI = (bi.z - bi.x) * (bi.w - bi.y);
            for (int j = i + 1 + tid; j < NPRE; j += blockDim.x) {
                float4 bj = sbox[j];
                float h = fmaxf(fminf(bi.z, bj.z) - fmaxf(bi.x, bj.x), 0.f);
                float w = fmaxf(fminf(bi.w, bj.w) - fmaxf(bi.y, bj.y), 0.f);
                float inter = h * w;
                float areaJ = (bj.z - bj.x) * (bj.w - bj.y);
                float iou = inter / (areaI + areaJ - inter + 1e-9f);
                if (iou > 0.7f) atomicAnd(&keepBits[j >> 5], ~(1u << (j & 31)));
            }
        }
        __syncthreads();
    }
    for (int r = tid; r < NPRE; r += blockDim.x)
        keepArr[r] = (keepBits[r >> 5] >> (r & 31)) & 1u;
}

// ---------------- kernel 7: replicate top_k(fs,300) selection ---------------
__global__ void k_finalize(const float4* __restrict__ bk, const unsigned* __restrict__ keepArr,
                           const float* __restrict__ topscore, float* __restrict__ rois) {
    if (threadIdx.x != 0 || blockIdx.x != 0) return;
    int cnt = 0;
    for (int r = 0; r < NPRE && cnt < NPOST; ++r) {
        if (keepArr[r] && topscore[r] != -INFINITY) {
            float4 b = bk[r];
            rois[cnt * 4 + 0] = b.x; rois[cnt * 4 + 1] = b.y;
            rois[cnt * 4 + 2] = b.z; rois[cnt * 4 + 3] = b.w;
            ++cnt;
        }
    }
    for (int r = 0; r < NPRE && cnt < NPOST; ++r) {
        if (!(keepArr[r] && topscore[r] != -INFINITY)) {
            float4 b = bk[r];
            rois[cnt * 4 + 0] = b.x; rois[cnt * 4 + 1] = b.y;
            rois[cnt * 4 + 2] = b.z; rois[cnt * 4 + 3] = b.w;
            ++cnt;
        }
    }
}

extern "C" void kernel_launch(void* const* d_in, const int* in_sizes, int n_in,
                              void* d_out, int out_size, void* d_ws, size_t ws_size,
                              hipStream_t stream) {
    (void)in_sizes; (void)n_in; (void)out_size; (void)ws_size;
    const float* feat    = (const float*)d_in[0];
    const float* conv1_w = (const float*)d_in[1];
    const float* conv1_b = (const float*)d_in[2];
    const float* score_w = (const float*)d_in[3];
    const float* score_b = (const float*)d_in[4];
    const float* loc_w   = (const float*)d_in[5];
    const float* loc_b   = (const float*)d_in[6];
    const int*   imh     = (const int*)d_in[7];
    const int*   imw     = (const int*)d_in[8];

    float* out = (float*)d_out;
    char*  ws  = (char*)d_ws;
    float*              wt      = (float*)(ws + WT_OFF);
    float*              xbuf    = (float*)(ws + X_OFF);
    float4*             boxes   = (float4*)(ws + BOX_OFF);
    unsigned long long* keys    = (unsigned long long*)(ws + KEY_OFF);
    float4*             bk      = (float4*)(ws + BK_OFF);
    float*              tscore  = (float*)(ws + TS_OFF);
    unsigned*           keepArr = (unsigned*)(ws + KEEP_OFF);

    k_transpose_w<<<(9 * CIN * CIN + 255) / 256, 256, 0, stream>>>(conv1_w, wt);
    k_conv3x3_wmma<<<dim3(30, 8), 256, 0, stream>>>(feat, wt, conv1_b, xbuf);
    k_conv1x1<<<MTOT, 64, 0, stream>>>(xbuf, loc_w, loc_b, score_w, score_b, out);
    k_prep<<<NSORT / 256, 256, 0, stream>>>(out + OFF_LOCS, out + OFF_SCORES, imh, imw,
                                            out + OFF_ANCHOR, boxes, keys);
    for (int k = 2; k <= NSORT; k <<= 1)
        for (int j = k >> 1; j > 0; j >>= 1)
            k_bitonic<<<NSORT / 256, 256, 0, stream>>>(keys, j, k);
    k_gather_top<<<(NPRE + 255) / 256, 256, 0, stream>>>(keys, boxes, bk, tscore, keepArr);
    k_nms<<<1, 1024, NPRE * sizeof(float4), stream>>>(bk, keepArr);
    k_finalize<<<1, 1, 0, stream>>>(bk, keepArr, tscore, out + OFF_ROIS);
}